// GVAE_77807627535080
// MI455X (gfx1250) — compile-verified
//
#include <hip/hip_runtime.h>
#include <hip/hip_bf16.h>

typedef float v2f __attribute__((ext_vector_type(2)));
typedef float v8f __attribute__((ext_vector_type(8)));

#define TPB 256
#define KB  64          // k-panel staged in LDS per iteration

// ---------------- degree / normalization ----------------

__global__ void deg_init_kernel(float* __restrict__ deg, int n) {
    int i = blockIdx.x * blockDim.x + threadIdx.x;
    if (i < n) deg[i] = 1.0f;   // self-loop contribution
}

__global__ void deg_scatter_kernel(const int* __restrict__ dst, float* __restrict__ deg, int e) {
    int i = blockIdx.x * blockDim.x + threadIdx.x;
    if (i < e) atomicAdd(&deg[dst[i]], 1.0f);
}

__global__ void deg_rsqrt_kernel(float* __restrict__ deg, int n) {
    int i = blockIdx.x * blockDim.x + threadIdx.x;
    if (i < n) {
        float d = deg[i];
        deg[i] = d > 0.0f ? rsqrtf(d) : 0.0f;
    }
}

// ---------------- fp32 WMMA GEMM: Y[M, NT*16] = X[M,K] @ W[K, NT*16] --------
// Block = 8 waves, owns 8 m-tiles and ALL NT n-tiles (A read exactly once per
// GEMM).  W k-panel staged in LDS, laid out linear in (k-pair, n-tile) so that
// every B fragment is one ds_load_b64 at a compile-time immediate offset from
// a single per-lane base (no per-step address math, no address-reg table).
//
// LDS element index for W[k][t*16+c15] = ((k>>1)*NT + t)*32 + c15*2 + (k&1)
// Wave-side read for WMMA step kk = k + half*2:
//   base  = l15*2 + half*NT*32                       (one VGPR, computed once)
//   imm   = ((k>>1)*NT + t)*32 floats                (compile-time constant)
//
// A 16x4 layout  : lanes 0-15 = rows, vgpr0/1 = K0/K1 (lanes 16-31 -> K2/K3)
// B 4x16 layout  : lanes 0-15 = cols, vgpr0/1 = K0/K1 (lanes 16-31 -> K2/K3)
// C/D 16x16      : vgpr r -> row r (lanes 0-15) / row r+8 (lanes 16-31)

template <int NT, bool RELU>
__global__ __launch_bounds__(TPB, 1)
void gemm_wmma_f32_kernel(const float* __restrict__ X,
                          const float* __restrict__ W,
                          float* __restrict__ Y,
                          int M, int K) {
    constexpr int Nd = NT * 16;
    __shared__ float lds_b[KB * Nd];             // up to 32 KB (NT=8)

    const int lane = threadIdx.x & 31;
    const int wave = threadIdx.x >> 5;
    const int mtiles = M >> 4;
    const int mtile  = blockIdx.x * 8 + wave;
    const bool active = (mtile < mtiles);        // wave-uniform guard

    const int half = lane >> 4;                  // 0: K pair {0,1}, 1: {2,3}
    const int l15  = lane & 15;
    const int arow = mtile * 16 + l15;
    const float* __restrict__ Xr = X + (size_t)(active ? arow : 0) * K;

    // per-lane LDS base: l15*2 floats + half * (NT*32) floats
    const float* __restrict__ Bp = lds_b + (l15 << 1) + half * (NT << 5);

    v8f c[NT];
#pragma unroll
    for (int t = 0; t < NT; ++t) c[t] = (v8f){};

    for (int kb = 0; kb < K; kb += KB) {
        // ---- stage W[kb:kb+KB, 0:Nd] into LDS (all 256 threads) ----
#pragma unroll
        for (int i = 0; i < (KB * Nd) / TPB; ++i) {
            int idx = threadIdx.x + i * TPB;     // 0 .. KB*Nd-1
            int k   = idx / Nd;
            int col = idx - k * Nd;
            int t   = col >> 4;
            int c15 = col & 15;
            lds_b[(((k >> 1) * NT + t) << 5) + (c15 << 1) + (k & 1)] =
                W[(size_t)(kb + k) * Nd + col];
        }
        __syncthreads();

        if (active) {
            if (kb + KB < K) __builtin_prefetch(Xr + kb + KB, 0, 3);
            const float* __restrict__ Ap = Xr + kb + (half << 1);
#pragma unroll
            for (int k = 0; k < KB; k += 4) {
                v2f a = *(const v2f*)(Ap + k);
                if (RELU) { a.x = fmaxf(a.x, 0.0f); a.y = fmaxf(a.y, 0.0f); }
#pragma unroll
                for (int t = 0; t < NT; ++t) {
                    v2f b = *(const v2f*)(Bp + ((((k >> 1) * NT + t)) << 5));
                    c[t] = __builtin_amdgcn_wmma_f32_16x16x4_f32(
                            /*neg_a=*/false, a, /*neg_b=*/false, b,
                            /*c_mod=*/(short)0, c[t],
                            /*reuse_a=*/false, /*reuse_b=*/false);
                }
            }
        }
        __syncthreads();
    }

    if (active) {
#pragma unroll
        for (int r = 0; r < 8; ++r) {
            const int row = mtile * 16 + r + half * 8;
#pragma unroll
            for (int t = 0; t < NT; ++t) {
                Y[(size_t)row * Nd + t * 16 + l15] = c[t][r];
            }
        }
    }
}

// ---------------- GCN aggregation ----------------
// Agg[i,f] = Hlin[i,f]*dinv[i]^2 + bias[f]   (self-loop term + bias, also init)
__global__ void selfloop_init_kernel(const float* __restrict__ Hlin,
                                     const float* __restrict__ dinv,
                                     const float* __restrict__ bias,
                                     float* __restrict__ Agg,
                                     int n, int F) {
    long long t = (long long)blockIdx.x * blockDim.x + threadIdx.x;
    long long total = (long long)n * F;
    if (t >= total) return;
    int node = (int)(t / F);
    int f    = (int)(t - (long long)node * F);
    float di = dinv[node];
    Agg[t] = Hlin[t] * di * di + bias[f];
}

// Agg[dst,f..f+3] += Hlin[src,f..f+3] * dinv[src]*dinv[dst]
__global__ void edge_scatter_kernel(const float* __restrict__ Hlin,
                                    const int* __restrict__ src,
                                    const int* __restrict__ dst,
                                    const float* __restrict__ dinv,
                                    float* __restrict__ Agg,
                                    int F, int e) {
    long long t = (long long)blockIdx.x * blockDim.x + threadIdx.x;
    const int chunks = F >> 2;                   // float4 chunks per edge
    long long total = (long long)e * chunks;
    if (t >= total) return;
    int ed = (int)(t / chunks);
    int f  = ((int)(t - (long long)ed * chunks)) << 2;
    int s  = src[ed];
    int d  = dst[ed];
    float norm = dinv[s] * dinv[d];
    const float4 h = *(const float4*)(Hlin + (size_t)s * F + f);
    float* out = Agg + (size_t)d * F + f;
    atomicAdd(out + 0, h.x * norm);
    atomicAdd(out + 1, h.y * norm);
    atomicAdd(out + 2, h.z * norm);
    atomicAdd(out + 3, h.w * norm);
}

// ---------------- reparameterization ----------------
__global__ void reparam_kernel(const float* __restrict__ mu,
                               const float* __restrict__ ls,
                               const float* __restrict__ eps,
                               float* __restrict__ z, long long n) {
    long long i = (long long)blockIdx.x * blockDim.x + threadIdx.x;
    if (i < n) z[i] = mu[i] + eps[i] * expf(ls[i]);
}

// ---------------- host orchestration ----------------

static inline int ceil_div_ll(long long a, int b) { return (int)((a + b - 1) / b); }

extern "C" void kernel_launch(void* const* d_in, const int* in_sizes, int n_in,
                              void* d_out, int out_size, void* d_ws, size_t ws_size,
                              hipStream_t stream) {
    const int IN = 256, HID = 128, LAT = 64, OUT = 32;

    const float* x   = (const float*)d_in[0];
    const int*   ei  = (const int*)  d_in[1];
    const float* W1  = (const float*)d_in[2];
    const float* b1  = (const float*)d_in[3];
    const float* W2  = (const float*)d_in[4];
    const float* b2  = (const float*)d_in[5];
    const float* Wmu = (const float*)d_in[6];
    const float* bmu = (const float*)d_in[7];
    const float* Wls = (const float*)d_in[8];
    const float* bls = (const float*)d_in[9];
    const float* eps = (const float*)d_in[10];

    const int N = in_sizes[0] / IN;
    const int E = in_sizes[1] / 2;
    const int* src = ei;
    const int* dst = ei + E;

    // workspace: dinv | bufA(N*128) | bufB(N*128)
    float* dinv = (float*)d_ws;
    size_t offA = (((size_t)N * sizeof(float)) + 255) & ~(size_t)255;
    float* bufA = (float*)((char*)d_ws + offA);
    float* bufB = bufA + (size_t)N * HID;

    float* out_z  = (float*)d_out;
    float* out_mu = out_z  + (size_t)N * OUT;
    float* out_ls = out_mu + (size_t)N * OUT;

    // 1) normalization coefficients
    deg_init_kernel<<<ceil_div_ll(N, TPB), TPB, 0, stream>>>(dinv, N);
    deg_scatter_kernel<<<ceil_div_ll(E, TPB), TPB, 0, stream>>>(dst, dinv, E);
    deg_rsqrt_kernel<<<ceil_div_ll(N, TPB), TPB, 0, stream>>>(dinv, N);

    const int gemm_blocks = ceil_div_ll(N / 16, 8);

    // 2) layer 1: h1 = relu(agg(x@W1) + b1)   (relu deferred into next GEMM)
    gemm_wmma_f32_kernel<8, false><<<gemm_blocks, TPB, 0, stream>>>(x, W1, bufA, N, IN);
    selfloop_init_kernel<<<ceil_div_ll((long long)N * HID, TPB), TPB, 0, stream>>>(bufA, dinv, b1, bufB, N, HID);
    edge_scatter_kernel<<<ceil_div_ll((long long)E * (HID / 4), TPB), TPB, 0, stream>>>(bufA, src, dst, dinv, bufB, HID, E);

    // 3) layer 2: h2 = agg(relu(h1)@W2) + b2
    gemm_wmma_f32_kernel<4, true><<<gemm_blocks, TPB, 0, stream>>>(bufB, W2, bufA, N, HID);
    selfloop_init_kernel<<<ceil_div_ll((long long)N * LAT, TPB), TPB, 0, stream>>>(bufA, dinv, b2, bufB, N, LAT);
    edge_scatter_kernel<<<ceil_div_ll((long long)E * (LAT / 4), TPB), TPB, 0, stream>>>(bufA, src, dst, dinv, bufB, LAT, E);

    // 4) mu / log_std heads (aggregate straight into d_out slices)
    float* mu_lin = bufA;                       // N*32
    float* ls_lin = bufA + (size_t)N * OUT;     // N*32
    gemm_wmma_f32_kernel<2, false><<<gemm_blocks, TPB, 0, stream>>>(bufB, Wmu, mu_lin, N, LAT);
    gemm_wmma_f32_kernel<2, false><<<gemm_blocks, TPB, 0, stream>>>(bufB, Wls, ls_lin, N, LAT);

    selfloop_init_kernel<<<ceil_div_ll((long long)N * OUT, TPB), TPB, 0, stream>>>(mu_lin, dinv, bmu, out_mu, N, OUT);
    edge_scatter_kernel<<<ceil_div_ll((long long)E * (OUT / 4), TPB), TPB, 0, stream>>>(mu_lin, src, dst, dinv, out_mu, OUT, E);

    selfloop_init_kernel<<<ceil_div_ll((long long)N * OUT, TPB), TPB, 0, stream>>>(ls_lin, dinv, bls, out_ls, N, OUT);
    edge_scatter_kernel<<<ceil_div_ll((long long)E * (OUT / 4), TPB), TPB, 0, stream>>>(ls_lin, src, dst, dinv, out_ls, OUT, E);

    // 5) z = mu + eps * exp(log_std)
    reparam_kernel<<<ceil_div_ll((long long)N * OUT, TPB), TPB, 0, stream>>>(out_mu, out_ls, eps, out_z, (long long)N * OUT);
}